// EMDLoss_18038862643365
// MI455X (gfx1250) — compile-verified
//
#include <hip/hip_runtime.h>
#include <cmath>
#include <cfloat>

typedef float v2f __attribute__((ext_vector_type(2)));
typedef float v8f __attribute__((ext_vector_type(8)));

#define SINK_BLUR 0.001f
#define TCOLS 256
#define LOG2E 1.44269504088896340736f
#define LN2 0.69314718055994530942f

#if defined(__has_builtin)
#if __has_builtin(__builtin_amdgcn_wmma_f32_16x16x4_f32)
#define USE_WMMA 1
#else
#define USE_WMMA 0
#endif
#if __has_builtin(__builtin_amdgcn_sqrtf)
#define FAST_SQRT(x) __builtin_amdgcn_sqrtf(x)
#else
#define FAST_SQRT(x) sqrtf(x)
#endif
#if __has_builtin(__builtin_amdgcn_exp2f)
#define FAST_EXP2(x) __builtin_amdgcn_exp2f(x)
#else
#define FAST_EXP2(x) exp2f(x)
#endif
#if __has_builtin(__builtin_amdgcn_logf)
#define FAST_LOG2(x) __builtin_amdgcn_logf(x)   /* v_log_f32 = log base 2 */
#else
#define FAST_LOG2(x) log2f(x)
#endif
#else
#define USE_WMMA 0
#define FAST_SQRT(x) sqrtf(x)
#define FAST_EXP2(x) exp2f(x)
#define FAST_LOG2(x) log2f(x)
#endif

// ---------------------------------------------------------------------------
// Pack point coords + squared norm into float4 (x,y,z,|p|^2)
// ---------------------------------------------------------------------------
__global__ void pack_norm_kernel(const float* __restrict__ src,
                                 float4* __restrict__ dst, int npts) {
  int p = blockIdx.x * blockDim.x + threadIdx.x;
  if (p < npts) {
    float a = src[3 * p + 0];
    float b = src[3 * p + 1];
    float c = src[3 * p + 2];
    dst[p] = make_float4(a, b, c, a * a + b * b + c * c);
  }
}

// ---------------------------------------------------------------------------
// diameter = max over dims of (max - min) over all points of x and y
// single block of 1024 threads
// ---------------------------------------------------------------------------
__global__ void diam_kernel(const float* __restrict__ x,
                            const float* __restrict__ y,
                            float* __restrict__ diam, int npts) {
  __shared__ float red[1024 * 6];
  int t = threadIdx.x;
  float mn0 = FLT_MAX, mn1 = FLT_MAX, mn2 = FLT_MAX;
  float mx0 = -FLT_MAX, mx1 = -FLT_MAX, mx2 = -FLT_MAX;
  for (int p = t; p < npts; p += blockDim.x) {
    float a = x[3 * p + 0], b = x[3 * p + 1], c = x[3 * p + 2];
    mn0 = fminf(mn0, a); mx0 = fmaxf(mx0, a);
    mn1 = fminf(mn1, b); mx1 = fmaxf(mx1, b);
    mn2 = fminf(mn2, c); mx2 = fmaxf(mx2, c);
    a = y[3 * p + 0]; b = y[3 * p + 1]; c = y[3 * p + 2];
    mn0 = fminf(mn0, a); mx0 = fmaxf(mx0, a);
    mn1 = fminf(mn1, b); mx1 = fmaxf(mx1, b);
    mn2 = fminf(mn2, c); mx2 = fmaxf(mx2, c);
  }
  red[6 * t + 0] = mn0; red[6 * t + 1] = mn1; red[6 * t + 2] = mn2;
  red[6 * t + 3] = mx0; red[6 * t + 4] = mx1; red[6 * t + 5] = mx2;
  __syncthreads();
  for (int off = 512; off > 0; off >>= 1) {
    if (t < off) {
      red[6 * t + 0] = fminf(red[6 * t + 0], red[6 * (t + off) + 0]);
      red[6 * t + 1] = fminf(red[6 * t + 1], red[6 * (t + off) + 1]);
      red[6 * t + 2] = fminf(red[6 * t + 2], red[6 * (t + off) + 2]);
      red[6 * t + 3] = fmaxf(red[6 * t + 3], red[6 * (t + off) + 3]);
      red[6 * t + 4] = fmaxf(red[6 * t + 4], red[6 * (t + off) + 4]);
      red[6 * t + 5] = fmaxf(red[6 * t + 5], red[6 * (t + off) + 5]);
    }
    __syncthreads();
  }
  if (t == 0) {
    float d0 = red[3] - red[0];
    float d1 = red[4] - red[1];
    float d2 = red[5] - red[2];
    diam[0] = fmaxf(d0, fmaxf(d1, d2));
  }
}

// ---------------------------------------------------------------------------
// Fused softmin pass (both directions, both batches, one launch).
// Online logsumexp runs in base-2 domain (v_exp_f32 = 2^x used directly).
// The WMMA accumulator carries C[m][n] = -x2_m/2, so the matrix op yields
//   d = x.y - x2/2   and   ||x-y||^2 = fma(-2, d, y2)  -- one VALU op.
//   mode 0: init      ->  out = -eps*LSE_j(log_w - C/eps)   (eps = max(diam*sc, blur))
//   mode 1: step      ->  out = 0.5*(old + softmin)
//   mode 2: final     ->  out = softmin at eps = blur
// One wave computes 16 rows; 16x16 dot-product tiles via V_WMMA_F32_16X16X4_F32
// (d=3 padded to K=4). 256 threads = 8 waves = 128 rows/block.
// ---------------------------------------------------------------------------
__global__ void __launch_bounds__(256)
sinkhorn_softmin_kernel(const float4* __restrict__ px,
                        const float4* __restrict__ py,
                        const float* __restrict__ f_old,
                        const float* __restrict__ g_old,
                        float* __restrict__ f_new,
                        float* __restrict__ g_new,
                        const float* __restrict__ diam_ptr,
                        float sc, float log_w, int N, int mode) {
  __shared__ float4 tP[TCOLS];
  __shared__ float tH[TCOLS];   // h * log2e per column

  const int tid = threadIdx.x;
  const int lane = tid & 31;
  const int wave = tid >> 5;
  const int l16 = lane & 15;
  const int hi = (lane >> 4) * 8;   // 0 for lanes 0-15, 8 for lanes 16-31

  const int bpd = N / 128;          // row-tiles per (dir,batch)
  int idx = blockIdx.x;
  int dir = idx / (2 * bpd);
  int rem = idx - dir * (2 * bpd);
  int b = rem / bpd;
  int tile = rem - b * bpd;

  const float4* rowP = dir ? py : px;
  const float4* colP = dir ? px : py;
  const float* hsrc = dir ? f_old : g_old;   // opposite potential for h
  const float* oldv = dir ? g_old : f_old;   // same-side old for averaging
  float* outv = dir ? g_new : f_new;

  const int gb = b * N;
  const float eps = (mode == 2) ? SINK_BLUR : fmaxf(diam_ptr[0] * sc, SINK_BLUR);
  const float inv_eps = 1.0f / eps;
  const float inv_eps2 = inv_eps * LOG2E;    // C scale into base-2 domain
  const float log_w2 = log_w * LOG2E;

  const int rowbase = tile * 128 + wave * 16;

  float x2r[8];
#if !USE_WMMA
  float xr0[8], xr1[8], xr2[8];
#endif
#pragma unroll
  for (int r = 0; r < 8; ++r) {
    float4 Pr = rowP[gb + rowbase + hi + r];
    x2r[r] = Pr.w;
#if !USE_WMMA
    xr0[r] = Pr.x; xr1[r] = Pr.y; xr2[r] = Pr.z;
#endif
  }

#if USE_WMMA
  // A fragment: 16x4 f32, lane<16 holds K=0,1 of row l16; lane>=16 holds K=2,3
  float4 P = rowP[gb + rowbase + l16];
  v2f afrag;
  afrag[0] = (lane < 16) ? P.x : P.z;
  afrag[1] = (lane < 16) ? P.y : 0.0f;   // K=3 pad
  // accumulator input: C[m][n] = -x2_m / 2 (row-constant, column-independent)
  v8f cacc;
#pragma unroll
  for (int r = 0; r < 8; ++r) cacc[r] = -0.5f * x2r[r];
#endif

  float m2[8], sv[8];
#pragma unroll
  for (int r = 0; r < 8; ++r) { m2[r] = -3.0e38f; sv[r] = 0.0f; }

  for (int c0 = 0; c0 < N; c0 += TCOLS) {
    __syncthreads();
    int j = c0 + tid;
    tP[tid] = colP[gb + j];
    tH[tid] = (mode == 0) ? log_w2 : fmaf(hsrc[gb + j], inv_eps2, log_w2);
    __syncthreads();

#pragma unroll 4
    for (int cc = 0; cc < TCOLS; cc += 16) {
      float4 Q = tP[cc + l16];
      float h2 = tH[cc + l16];
      float y2 = Q.w;
#if USE_WMMA
      v2f bfrag;
      bfrag[0] = (lane < 16) ? Q.x : Q.z;
      bfrag[1] = (lane < 16) ? Q.y : 0.0f;
      // d[r] = x_{rowbase+hi+r} . y_{c0+cc+l16} - x2_r/2
      v8f d = __builtin_amdgcn_wmma_f32_16x16x4_f32(
          false, afrag, false, bfrag, (short)0, cacc, false, false);
#endif
#pragma unroll
      for (int r = 0; r < 8; ++r) {
#if USE_WMMA
        float dp = d[r];
#else
        float dp = fmaf(xr0[r], Q.x,
                   fmaf(xr1[r], Q.y,
                   fmaf(xr2[r], Q.z, -0.5f * x2r[r])));
#endif
        float sq = fmaf(-2.0f, dp, y2);        // x2 + y2 - 2*x.y
        sq = fmaxf(sq, 1e-12f);
        float val2 = fmaf(-inv_eps2, FAST_SQRT(sq), h2);  // (h - C/eps)*log2e
        float nm = fmaxf(m2[r], val2);
        sv[r] = fmaf(sv[r], FAST_EXP2(m2[r] - nm), FAST_EXP2(val2 - nm));
        m2[r] = nm;
      }
    }
  }

  // combine (m2,s) across the 16 lanes of each half-wave (butterfly)
#pragma unroll
  for (int mask = 1; mask < 16; mask <<= 1) {
#pragma unroll
    for (int r = 0; r < 8; ++r) {
      float om = __shfl_xor(m2[r], mask, 32);
      float os = __shfl_xor(sv[r], mask, 32);
      float nm = fmaxf(m2[r], om);
      sv[r] = fmaf(sv[r], FAST_EXP2(m2[r] - nm), os * FAST_EXP2(om - nm));
      m2[r] = nm;
    }
  }

  if (l16 == 0) {  // lane 0 writes rows 0-7, lane 16 writes rows 8-15
    int rb = gb + rowbase + hi;
    const float neps_ln2 = -eps * LN2;
#pragma unroll
    for (int r = 0; r < 8; ++r) {
      // -eps*(m + ln s) = -eps*ln2*(m2 + log2 s)
      float ft = neps_ln2 * (m2[r] + FAST_LOG2(sv[r]));
      float res = (mode == 1) ? 0.5f * (oldv[rb + r] + ft) : ft;
      outv[rb + r] = res;
    }
  }
}

// ---------------------------------------------------------------------------
// loss = (sum f_fin + sum g_fin) / (B*N)    (a=1/N, b=1/M, mean over B; N==M)
// ---------------------------------------------------------------------------
__global__ void loss_reduce_kernel(const float* __restrict__ ff,
                                   const float* __restrict__ gf,
                                   float* __restrict__ out, int total,
                                   float scale) {
  __shared__ float red[1024];
  int t = threadIdx.x;
  float s = 0.0f;
  for (int i = t; i < total; i += blockDim.x) s += ff[i] + gf[i];
  red[t] = s;
  __syncthreads();
  for (int off = 512; off > 0; off >>= 1) {
    if (t < off) red[t] += red[t + off];
    __syncthreads();
  }
  if (t == 0) out[0] = red[0] * scale;
}

// ---------------------------------------------------------------------------
extern "C" void kernel_launch(void* const* d_in, const int* in_sizes, int n_in,
                              void* d_out, int out_size, void* d_ws,
                              size_t ws_size, hipStream_t stream) {
  const float* x = (const float*)d_in[0];
  const float* y = (const float*)d_in[1];
  float* out = (float*)d_out;

  const int B = 2, D = 3;
  const int N = in_sizes[0] / (B * D);  // 8192
  const int npts = B * N;

  // workspace carving (256B aligned)
  char* ws = (char*)d_ws;
  size_t off = 0;
  auto carve = [&](size_t bytes) -> void* {
    void* p = (void*)(ws + off);
    off = (off + bytes + 255) & ~(size_t)255;
    return p;
  };
  float4* px = (float4*)carve((size_t)npts * sizeof(float4));
  float4* py = (float4*)carve((size_t)npts * sizeof(float4));
  float* f0 = (float*)carve((size_t)npts * sizeof(float));
  float* f1 = (float*)carve((size_t)npts * sizeof(float));
  float* g0 = (float*)carve((size_t)npts * sizeof(float));
  float* g1 = (float*)carve((size_t)npts * sizeof(float));
  float* ffin = (float*)carve((size_t)npts * sizeof(float));
  float* gfin = (float*)carve((size_t)npts * sizeof(float));
  float* diam = (float*)carve(256);

  const int thr = 256;
  pack_norm_kernel<<<(npts + thr - 1) / thr, thr, 0, stream>>>(x, px, npts);
  pack_norm_kernel<<<(npts + thr - 1) / thr, thr, 0, stream>>>(y, py, npts);
  diam_kernel<<<1, 1024, 0, stream>>>(x, y, diam, npts);

  const float log_w = -logf((float)N);
  const int grid = 2 * B * (N / 128);

  // init at eps_list[0] = max(diam, blur): writes f0, g0 (old buffers unread)
  sinkhorn_softmin_kernel<<<grid, 256, 0, stream>>>(
      px, py, f1, g1, f0, g0, diam, 1.0f, log_w, N, 0);

  // 96 annealed symmetric half-steps, double-buffered
  float* fc = f0; float* gc = g0;
  float* fa = f1; float* ga = g1;
  double sc = 1.0;
  for (int k = 0; k < 96; ++k) {
    sinkhorn_softmin_kernel<<<grid, 256, 0, stream>>>(
        px, py, fc, gc, fa, ga, diam, (float)sc, log_w, N, 1);
    float* t;
    t = fc; fc = fa; fa = t;
    t = gc; gc = ga; ga = t;
    sc *= 0.9;
  }

  // final extrapolation at eps = blur
  sinkhorn_softmin_kernel<<<grid, 256, 0, stream>>>(
      px, py, fc, gc, ffin, gfin, diam, 0.0f, log_w, N, 2);

  loss_reduce_kernel<<<1, 1024, 0, stream>>>(ffin, gfin, out, npts,
                                             1.0f / (float)npts);
}